// dggcn_sparse_assemble_32006096290285
// MI455X (gfx1250) — compile-verified
//
#include <hip/hip_runtime.h>

typedef __attribute__((ext_vector_type(2))) float v2f;
typedef __attribute__((ext_vector_type(8))) float v8f;
typedef __attribute__((ext_vector_type(4))) unsigned int u32x4;
typedef __attribute__((ext_vector_type(8))) int i32x8;
typedef __attribute__((ext_vector_type(4))) int i32x4;

#define Bn 128
#define Cn 64
#define Tn 256
#define Vn 25
#define Sn 3
#define On 64
#define KFLAT (Sn * Cn)      // 192
#define TT 2
#define NCOL (TT * Vn)       // 50 columns per block tile

// ---------------------------------------------------------------------------
// Pass 1: y_pre[b,o,t,w] = sum_{s,c} W[s,o,c] * (sum_v x[b,c,t,v]*A[s,v,w])
//   Stage 0: TDM (tensor_load_to_lds) stages X tile 64x50 into LDS
//   Stage A: XA = X @ A[s] per (s,tt) via f32 WMMA (K=25 masked to 7x4)
//   Stage B: Y(64x50) = Wflat(64x192) @ XA(192x50) via f32 WMMA
// Fuses per-channel BN statistics via shuffle reduction + LDS/global atomics.
// ---------------------------------------------------------------------------
__global__ __launch_bounds__(256) void dggcn_pass1(
    const float* __restrict__ x, const float* __restrict__ A,
    const float* __restrict__ W, float* __restrict__ y,
    float* __restrict__ stats) {
  __shared__ float Xs[Cn][NCOL];         // 12.8 KB (TDM packs rows contiguously)
  __shared__ float As[Sn][Vn][Vn];       // 7.5 KB
  __shared__ float XAs[KFLAT][NCOL];     // 38.4 KB
  __shared__ float shs[On], shq[On];     // BN partial stats

  const int tid = threadIdx.x;
  const int b  = blockIdx.y;
  const int t0 = blockIdx.x * TT;

  // ---- Stage 0: TDM async load of X tile (rows c=0..63, 50 contiguous f32,
  //      row stride T*V=6400 elements) into Xs. Issued by wave 0 only.
  if (tid < 32) {
    unsigned long long ga = (unsigned long long)(const void*)x
                          + ((unsigned long long)b * Cn * Tn * Vn + (unsigned long long)t0 * Vn) * 4ull;
    unsigned lds_off = (unsigned)(unsigned long long)(const void*)&Xs[0][0];

    u32x4 g0;
    g0.x = 1u;                                   // count=1, is_restore=0, gather off
    g0.y = lds_off;                              // bits 63:32  lds_addr
    g0.z = (unsigned)(ga & 0xFFFFFFFFull);       // bits 95:64  global_addr lo
    g0.w = (unsigned)(ga >> 32) | 0x80000000u;   // bits 120:96 addr hi | type=2<<30

    const unsigned N0 = 1u << 30;                // tensor_dim0 (large, no OOB)
    const unsigned N1 = Cn;                      // tensor_dim1 = 64 rows
    const unsigned TILE0 = NCOL;                 // 50 elements, contiguous
    const unsigned TILE1 = Cn;                   // 64 rows
    const unsigned STRIDE0 = Tn * Vn;            // 6400 elements between rows
    i32x8 g1;
    g1[0] = (int)(2u << 16);                     // data_size=2 (4 bytes)
    g1[1] = (int)((N0 & 0xFFFFu) << 16);         // barrier addr=0 | dim0 lo
    g1[2] = (int)((N0 >> 16) | ((N1 & 0xFFFFu) << 16));
    g1[3] = (int)((N1 >> 16) | (TILE0 << 16));   // dim1 hi | tile_dim0
    g1[4] = (int)(TILE1);                        // tile_dim1 | tile_dim2=0
    g1[5] = (int)(STRIDE0);                      // tensor_dim0_stride lo
    g1[6] = 0;                                   // stride0 hi | stride1 lo
    g1[7] = 0;                                   // stride1 hi
    i32x4 gz4 = {0, 0, 0, 0};
    i32x8 gz8 = {0, 0, 0, 0, 0, 0, 0, 0};
    __builtin_amdgcn_tensor_load_to_lds(g0, g1, gz4, gz4, gz8, 0);
    __builtin_amdgcn_s_wait_tensorcnt(0);
  }

  // ---- stage A matrices into LDS while TDM runs
  for (int i = tid; i < Sn * Vn * Vn; i += 256) (&As[0][0][0])[i] = A[i];
  if (tid < On) { shs[tid] = 0.f; shq[tid] = 0.f; }
  __syncthreads();

  const int wave = tid >> 5, lane = tid & 31;
  const int half = lane >> 4;      // frag halves: lanes 0-15 -> K{0,1}, 16-31 -> K{2,3}
  const int lr   = lane & 15;

  // ---- Stage A via WMMA: XA[(s,c)][tt*25+w] = sum_v Xs[c][tt*25+v]*A[s][v][w]
  // Wave owns fixed M-tile (wave>>1) and N-tile (wave&1); loops (s,tt) combos.
  // B fragments (A[s][v][w]) are tt-invariant: loaded once per s into registers.
  {
    const int mtA  = wave >> 1;
    const int ntA  = wave & 1;
    const int crow = mtA * 16 + lr;
    const int wA   = ntA * 16 + lr;
    const int wcA  = (wA < Vn) ? wA : 0;
    const bool wok = (wA < Vn);

    for (int s = 0; s < Sn; ++s) {
      // load+select B fragments for this s (unconditional LDS load, cndmask)
      float br[7][2];
#pragma unroll
      for (int kks = 0; kks < 7; ++kks) {
        int k0 = kks * 4 + half * 2;
        bool m0 = (k0 < Vn), m1 = (k0 + 1 < Vn);
        int k0a = m0 ? k0 : 0;
        int k1a = m1 ? k0 + 1 : 0;
        float b0 = As[s][k0a][wcA];          // always in-bounds (clamped)
        float b1 = As[s][k1a][wcA];
        br[kks][0] = (m0 & wok) ? b0 : 0.f;
        br[kks][1] = (m1 & wok) ? b1 : 0.f;
      }
#pragma unroll
      for (int tt = 0; tt < TT; ++tt) {
        v8f acc = {0.f, 0.f, 0.f, 0.f, 0.f, 0.f, 0.f, 0.f};
#pragma unroll
        for (int kks = 0; kks < 7; ++kks) {
          int k0 = kks * 4 + half * 2;
          bool m0 = (k0 < Vn), m1 = (k0 + 1 < Vn);
          int k0a = m0 ? k0 : 0;
          int k1a = m1 ? k0 + 1 : 0;
          float a0 = Xs[crow][tt * Vn + k0a];  // always in-bounds (clamped)
          float a1 = Xs[crow][tt * Vn + k1a];
          v2f a, bf;
          a.x  = m0 ? a0 : 0.f;
          a.y  = m1 ? a1 : 0.f;
          bf.x = br[kks][0];
          bf.y = br[kks][1];
          acc = __builtin_amdgcn_wmma_f32_16x16x4_f32(false, a, false, bf,
                                                      (short)0, acc, false, false);
        }
        if (wok) {
#pragma unroll
          for (int i = 0; i < 8; ++i)
            XAs[s * 64 + mtA * 16 + i + half * 8][tt * Vn + wA] = acc[i];
        }
      }
    }
  }
  __syncthreads();

  // ---- Stage B WMMA GEMM: Y(64 x 50) = Wflat(64 x 192) @ XA(192 x 50)
  // 8 waves: wave w owns M-tile (w&3) and N-tiles {2*(w>>2), 2*(w>>2)+1}
  const int m    = wave & 3;
  const int n0   = (wave >> 2) * 2;
  const int o_a  = m * 16 + lr;                 // A-matrix row (output channel)
  const int col0 = n0 * 16 + lr;                // B/D column for tile 0 (<=47)
  const int col1 = col0 + 16;                   // B/D column for tile 1
  const int cc1  = col1 < NCOL ? col1 : 0;

  v8f acc0 = {0.f, 0.f, 0.f, 0.f, 0.f, 0.f, 0.f, 0.f};
  v8f acc1 = {0.f, 0.f, 0.f, 0.f, 0.f, 0.f, 0.f, 0.f};

  for (int kk = 0; kk < KFLAT / 4; ++kk) {
    int k0 = kk * 4 + half * 2;              // this half-wave's K pair
    int s = k0 >> 6, c = k0 & 63;            // c even; c+1 stays inside branch s
    const float* wp = W + ((size_t)(s * On + o_a) * Cn + c);
    v2f a;  a.x = wp[0];          a.y = wp[1];
    v2f b0; b0.x = XAs[k0][col0]; b0.y = XAs[k0 + 1][col0];
    float b1x = XAs[k0][cc1], b1y = XAs[k0 + 1][cc1];
    v2f b1; b1.x = (col1 < NCOL) ? b1x : 0.f;
            b1.y = (col1 < NCOL) ? b1y : 0.f;
    acc0 = __builtin_amdgcn_wmma_f32_16x16x4_f32(false, a, false, b0,
                                                 (short)0, acc0, false, false);
    acc1 = __builtin_amdgcn_wmma_f32_16x16x4_f32(false, a, false, b1,
                                                 (short)0, acc1, false, false);
  }

  // ---- store D tiles + per-channel BN partials (butterfly shuffle reduce)
  float* yb = y + (size_t)b * On * Tn * Vn + (size_t)t0 * Vn;
#pragma unroll
  for (int i = 0; i < 8; ++i) {
    int o = m * 16 + i + half * 8;           // C/D layout: VGPR i -> rows i, i+8
    float v0 = acc0[i];
    float v1 = acc1[i];
    yb[(size_t)o * Tn * Vn + col0] = v0;     // col0 <= 47, always valid
    if (col1 < NCOL) yb[(size_t)o * Tn * Vn + col1] = v1;
    float v1m = (col1 < NCOL) ? v1 : 0.f;
    float p = v0 + v1m;
    float q = v0 * v0 + v1m * v1m;
#pragma unroll
    for (int d = 1; d < 16; d <<= 1) {       // reduce within 16-lane col group
      p += __shfl_xor(p, d, 32);
      q += __shfl_xor(q, d, 32);
    }
    if (lr == 0) {                           // lanes 0 and 16: two channels
      atomicAdd(&shs[o], p);
      atomicAdd(&shq[o], q);
    }
  }
  __syncthreads();
  if (tid < On) {
    atomicAdd(&stats[tid], shs[tid]);
    atomicAdd(&stats[On + tid], shq[tid]);
  }
}

// ---------------------------------------------------------------------------
// Pass 2: finalize BN affine params (scale, shift) per channel.
// ---------------------------------------------------------------------------
__global__ void dggcn_pass2(const float* __restrict__ stats,
                            const float* __restrict__ gamma,
                            const float* __restrict__ beta,
                            float* __restrict__ ab) {
  int o = threadIdx.x;
  if (o < On) {
    const float Ntot = (float)Bn * (float)Tn * (float)Vn;   // 819200
    float mean = stats[o] / Ntot;
    float var  = stats[On + o] / Ntot - mean * mean;
    float inv  = rsqrtf(var + 1e-5f);
    float sc   = gamma[o] * inv;
    ab[o]      = sc;
    ab[On + o] = beta[o] - mean * sc;
  }
}

// ---------------------------------------------------------------------------
// Pass 3: out = relu(y_pre*scale[o] + shift[o] + x), in place on d_out,
// vectorized float4 (T*V = 6400 divisible by 4 -> one channel per float4).
// ---------------------------------------------------------------------------
__global__ __launch_bounds__(256) void dggcn_pass3(
    float* __restrict__ y, const float* __restrict__ x,
    const float* __restrict__ ab) {
  const size_t n4 = (size_t)Bn * On * Tn * Vn / 4;
  size_t i = (size_t)blockIdx.x * blockDim.x + threadIdx.x;
  if (i >= n4) return;
  int o = (int)((i / (Tn * Vn / 4)) % On);   // 1600 float4s per (b,o) row
  float sc = ab[o], sh = ab[On + o];
  float4 yv = ((const float4*)y)[i];
  float4 xv = ((const float4*)x)[i];
  float4 r;
  r.x = fmaxf(fmaf(yv.x, sc, sh) + xv.x, 0.f);
  r.y = fmaxf(fmaf(yv.y, sc, sh) + xv.y, 0.f);
  r.z = fmaxf(fmaf(yv.z, sc, sh) + xv.z, 0.f);
  r.w = fmaxf(fmaf(yv.w, sc, sh) + xv.w, 0.f);
  ((float4*)y)[i] = r;
}

extern "C" void kernel_launch(void* const* d_in, const int* in_sizes, int n_in,
                              void* d_out, int out_size, void* d_ws, size_t ws_size,
                              hipStream_t stream) {
  const float* x     = (const float*)d_in[0];
  const float* A     = (const float*)d_in[1];
  const float* W     = (const float*)d_in[2];
  const float* gamma = (const float*)d_in[3];
  const float* beta  = (const float*)d_in[4];
  float* out   = (float*)d_out;
  float* stats = (float*)d_ws;        // 128 floats: sum | sumsq
  float* ab    = stats + 2 * On;      // 128 floats: scale | shift

  (void)hipMemsetAsync(stats, 0, 2 * On * sizeof(float), stream);

  dim3 g1(Tn / TT, Bn);
  dggcn_pass1<<<g1, 256, 0, stream>>>(x, A, W, out, stats);
  dggcn_pass2<<<1, 64, 0, stream>>>(stats, gamma, beta, ab);

  size_t n4 = (size_t)Bn * On * Tn * Vn / 4;
  dggcn_pass3<<<(unsigned)((n4 + 255) / 256), 256, 0, stream>>>(out, x, ab);
}